// VectorQuantiser_67972152427053
// MI455X (gfx1250) — compile-verified
//
#include <hip/hip_runtime.h>
#include <hip/hip_bf16.h>

// VQ-VAE vector quantiser for MI455X (gfx1250, wave32).
// dist = l2norm(z-rows) @ l2norm(codes)^T via V_WMMA_F32_16X16X4_F32.
// Codebook pre-normalized into d_ws, then streamed into double-buffered LDS
// tiles with GLOBAL_LOAD_ASYNC_TO_LDS_B128 (ASYNCcnt), overlapping the WMMAs.

#define K_CODES   8192
#define EDIM      64
#define HWSZ      4096            // H*W
#define N_ROWS    32768           // B*H*W
#define TILE_CODES 64             // codes staged into LDS per pipeline stage
#define LDS_STRIDE 68             // 68*4 = 272 B row pitch: 16B-aligned B128
                                  // targets, conflict-free (bank = (4j+c)%64),
                                  // 8B-aligned ds_load_b64 pair reads
#define BUF_FLOATS (TILE_CODES * LDS_STRIDE)

// d_out layout (floats): z_q[2097152] | loss[1] | idx[32768] | bincount[8192]
#define OFF_LOSS  2097152
#define OFF_IDX   2097153
#define OFF_BIN   2129921

typedef __attribute__((ext_vector_type(2))) float v2f;
typedef __attribute__((ext_vector_type(8))) float v8f;

// ---------------------------------------------------------------------------
// Kernel 1: write l2-normalized codebook into ws; zero loss + bincount slots.
// One wave32 per codebook row (64 floats -> 2 per lane).
// ---------------------------------------------------------------------------
__global__ __launch_bounds__(256) void vq_prep(const float* __restrict__ E,
                                               float* __restrict__ cn,
                                               float* __restrict__ out) {
    const int t = blockIdx.x * 256 + threadIdx.x;
    if (t < K_CODES) out[OFF_BIN + t] = 0.0f;
    if (t == 0)      out[OFF_LOSS]    = 0.0f;

    const int row  = t >> 5;
    const int lane = t & 31;
    if (row < K_CODES) {
        float x0 = E[row * EDIM + lane];
        float x1 = E[row * EDIM + lane + 32];
        float s  = x0 * x0 + x1 * x1;
        #pragma unroll
        for (int m = 16; m >= 1; m >>= 1) s += __shfl_xor(s, m);
        const float r = 1.0f / fmaxf(sqrtf(s), 1e-12f);
        cn[row * EDIM + lane]      = x0 * r;
        cn[row * EDIM + lane + 32] = x1 * r;
    }
}

// Issue one pipeline stage: copy 64 normalized codes (16 KB) into LDS buffer
// `buf` with async DMA (4 x B128 per thread, 256 threads). No wait here.
__device__ __forceinline__ void stage_async(const float* __restrict__ cn,
                                            int cb, unsigned lds_base_bytes,
                                            int t) {
    const float* gbase = cn + cb * EDIM;
    #pragma unroll
    for (int i = 0; i < 4; ++i) {
        const int e4  = (t + 256 * i) * 4;      // element offset (16B granules)
        const int row = e4 >> 6;
        const int col = e4 & 63;
        const float* g = gbase + e4;
        const unsigned l = lds_base_bytes + (unsigned)(row * LDS_STRIDE + col) * 4u;
        asm volatile("global_load_async_to_lds_b128 %0, %1, off"
                     :: "v"(l), "v"(g) : "memory");
    }
}

// ---------------------------------------------------------------------------
// Kernel 2: main. Each wave owns 16 z-rows (A in VGPRs); block of 8 waves
// sweeps the codebook through double-buffered async LDS tiles.
// ---------------------------------------------------------------------------
__global__ __launch_bounds__(256) void vq_main(const float* __restrict__ z,
                                               const float* __restrict__ E,
                                               const float* __restrict__ cn,
                                               float* __restrict__ out,
                                               float* __restrict__ partials) {
    __shared__ float lds[2 * BUF_FLOATS];
    __shared__ int   idx_s[8 * 16];

    const int t    = threadIdx.x;
    const int wave = t >> 5;
    const int lane = t & 31;
    const int j    = lane & 15;     // column / row-in-tile index
    const int hi   = lane >> 4;     // half-wave select

    const int tile  = blockIdx.x * 8 + wave;   // 0..2047
    const int n0    = tile * 16;               // first flattened row
    const int bimg  = n0 >> 12;                // n0 / (H*W)
    const int rem   = n0 & 4095;               // h*W + w0 (16 rows contiguous)
    const int zbase = bimg * (EDIM * HWSZ) + rem;   // + c*HWSZ + m

    const unsigned lds_base = (unsigned)(size_t)(&lds[0]);  // LDS byte offset

    // ---- A operand: 16x64 z tile, ISA 16x4-f32 layout, normalized rows ----
    v2f a[16];
    float ss = 0.0f;
    #pragma unroll
    for (int kk = 0; kk < 16; ++kk) {
        const int c0 = 4 * kk + 2 * hi;
        float x = z[zbase + c0 * HWSZ + j];
        float y = z[zbase + (c0 + 1) * HWSZ + j];
        a[kk].x = x; a[kk].y = y;
        ss += x * x + y * y;
    }
    ss += __shfl_xor(ss, 16);                       // lanes m / m+16 -> full row
    const float rinv = 1.0f / fmaxf(sqrtf(ss), 1e-12f);
    #pragma unroll
    for (int kk = 0; kk < 16; ++kk) { a[kk].x *= rinv; a[kk].y *= rinv; }

    float bestv[8];
    int   besti[8];
    #pragma unroll
    for (int r = 0; r < 8; ++r) { bestv[r] = -2.0f; besti[r] = 0; }

    // ---- prime the pipeline: stage codes [0,64) into buffer 0 ----
    stage_async(cn, 0, lds_base, t);
    int p = 0;

    for (int cb = 0; cb < K_CODES; cb += TILE_CODES) {
        asm volatile("s_wait_asynccnt 0" ::: "memory");  // my fills of buf p done
        __syncthreads();   // all waves' fills done; buf p^1 reads all retired

        if (cb + TILE_CODES < K_CODES)                   // prefetch next tile
            stage_async(cn, cb + TILE_CODES, lds_base + (p ^ 1) * (BUF_FLOATS * 4u), t);

        const int pbase = p * BUF_FLOATS;
        #pragma unroll
        for (int tt = 0; tt < 4; ++tt) {                 // 4 x 16-code WMMA tiles
            const int clb = tt * 16;
            v8f acc = {};
            #pragma unroll
            for (int kk = 0; kk < 16; ++kk) {            // K = 64 in steps of 4
                const int c0 = 4 * kk + 2 * hi;
                v2f bb;
                bb.x = lds[pbase + (clb + j) * LDS_STRIDE + c0];
                bb.y = lds[pbase + (clb + j) * LDS_STRIDE + c0 + 1];
                acc = __builtin_amdgcn_wmma_f32_16x16x4_f32(
                        /*neg_a=*/false, a[kk], /*neg_b=*/false, bb,
                        /*c_mod=*/(short)0, acc, /*reuse_a=*/false, /*reuse_b=*/false);
            }
            const int codeidx = cb + clb + j;
            #pragma unroll
            for (int r = 0; r < 8; ++r) {                // rows r + 8*hi
                const float v = acc[r];
                if (v > bestv[r]) { bestv[r] = v; besti[r] = codeidx; }
            }
        }
        p ^= 1;
    }

    // ---- argmax across the 16 lanes of each half (first-max tie-break) ----
    #pragma unroll
    for (int r = 0; r < 8; ++r) {
        float v = bestv[r];
        int   i = besti[r];
        #pragma unroll
        for (int m = 8; m >= 1; m >>= 1) {
            const float ov = __shfl_xor(v, m);
            const int   oi = __shfl_xor(i, m);
            if (ov > v || (ov == v && oi < i)) { v = ov; i = oi; }
        }
        bestv[r] = v; besti[r] = i;
    }

    if (j == 0) {
        #pragma unroll
        for (int r = 0; r < 8; ++r) {
            idx_s[wave * 16 + hi * 8 + r] = besti[r];
            atomicAdd(&out[OFF_BIN + besti[r]], 1.0f);   // exact +1.0 increments
        }
    }
    __syncthreads();

    const int myidx = idx_s[wave * 16 + j];
    if (hi == 0) out[OFF_IDX + n0 + j] = (float)myidx;

    // ---- gather z_q (== straight-through output), write, loss partial ----
    float acc2 = 0.0f;
    #pragma unroll
    for (int cc = 0; cc < 32; ++cc) {
        const int c    = hi * 32 + cc;
        const float q  = E[myidx * EDIM + c];
        const int addr = zbase + c * HWSZ + j;
        const float zv = z[addr];
        out[addr] = q;
        const float d = q - zv;
        acc2 += d * d;
    }
    #pragma unroll
    for (int m = 16; m >= 1; m >>= 1) acc2 += __shfl_xor(acc2, m);
    if (lane == 0) partials[tile] = acc2;
}

// ---------------------------------------------------------------------------
// Kernel 3: deterministic fixed-shape loss reduction of 2048 wave partials.
// loss = (1 + BETA) * mean((z_q - z)^2),  BETA = 0.25.
// ---------------------------------------------------------------------------
__global__ void vq_loss(const float* __restrict__ partials, float* __restrict__ out) {
    const int lane = threadIdx.x;       // 32 threads
    float s = 0.0f;
    for (int i = lane; i < 2048; i += 32) s += partials[i];
    #pragma unroll
    for (int m = 16; m >= 1; m >>= 1) s += __shfl_xor(s, m);
    if (lane == 0) out[OFF_LOSS] = s * (1.25f / 2097152.0f);
}

extern "C" void kernel_launch(void* const* d_in, const int* in_sizes, int n_in,
                              void* d_out, int out_size, void* d_ws, size_t ws_size,
                              hipStream_t stream) {
    (void)in_sizes; (void)n_in; (void)out_size; (void)ws_size;
    const float* z = (const float*)d_in[0];     // [8,64,64,64] f32
    const float* E = (const float*)d_in[1];     // [8192,64] f32
    float* out = (float*)d_out;

    float* cn_norm  = (float*)d_ws;                 // 8192*64 floats (2 MB)
    float* partials = cn_norm + K_CODES * EDIM;     // 2048 floats

    vq_prep<<<K_CODES / 8, 256, 0, stream>>>(E, cn_norm, out);
    vq_main<<<(N_ROWS / 16) / 8, 256, 0, stream>>>(z, E, cn_norm, out, partials);
    vq_loss<<<1, 32, 0, stream>>>(partials, out);
}